// ToneMapping_10746008174745
// MI455X (gfx1250) — compile-verified
//
#include <hip/hip_runtime.h>
#include <hip/hip_bf16.h>

typedef float v4f __attribute__((ext_vector_type(4)));

#define TPB 256
#define MAXB 64

// Rational-quadratic spline eval, bit-matching the JAX reference arithmetic:
//   t   = (xf - x_low) * invw               (invw = 1/((c+w)-c + 1e-8))
//   num = sl*t^2 + 2t(1-t)  = t * fma(sl-2, t, 2)
//   den = s*t^2 + 2(s-2)t+2 = fma(t, fma(s, t, 2s-4), 2)
//   y   = y_low + dy * num / den
__device__ __forceinline__ float spline_eval5(float xv, float e0, float e1, float e2,
                                              float e3, const v4f* __restrict__ s_bin) {
    float xf = fminf(fmaxf(xv, 0.0f), 1.0f);
    int idx = 0;
    idx += (e0 < xf) ? 1 : 0;      // searchsorted side='left': count edges < xf
    idx += (e1 < xf) ? 1 : 0;
    idx += (e2 < xf) ? 1 : 0;
    idx += (e3 < xf) ? 1 : 0;
    v4f b0 = s_bin[2 * idx];       // {x_low, invw, y_low, dy}   ds_load_b128
    v4f b1 = s_bin[2 * idx + 1];   // {sl-2,  s,    2s-4,  0}    ds_load_b128
    float t   = (xf - b0.x) * b0.y;
    float num = t * __builtin_fmaf(b1.x, t, 2.0f);
    float den = __builtin_fmaf(t, __builtin_fmaf(b1.y, t, b1.z), 2.0f);
    float r   = __builtin_amdgcn_rcpf(den);                      // v_rcp_f32
    r = __builtin_fmaf(__builtin_fmaf(-den, r, 1.0f), r, r);     // 1 Newton step
    return __builtin_fmaf(b0.w * num, r, b0.z);
}

// Hot kernel for num_bins == 5 (plain function, defined first so it leads the disasm).
__global__ __launch_bounds__(TPB) void tonemap5(
    const float* __restrict__ x, const float* __restrict__ widths,
    const float* __restrict__ heights, const float* __restrict__ slopes,
    float* __restrict__ out, int n) {
    __shared__ v4f   s_bin[10];
    __shared__ float s_edge[5];

    const int tid = threadIdx.x;
    if (tid < 5) {
        float c = 0.0f;
        for (int k = 0; k < tid; ++k) c += widths[k];   // widths_cumsum, same FP order as ref
        float w    = widths[tid];
        float xh   = c + w;                             // bin_edges[tid]
        float invw = 1.0f / ((xh - c) + 1e-8f);
        float yl   = heights[tid];
        float dy   = heights[tid + 1] - yl;
        float sl   = slopes[tid];
        float s    = sl + slopes[tid + 1];
        v4f b0 = {c, invw, yl, dy};
        v4f b1 = {sl - 2.0f, s, 2.0f * s - 4.0f, 0.0f};
        s_bin[2 * tid]     = b0;
        s_bin[2 * tid + 1] = b1;
        s_edge[tid] = xh;
    }
    __syncthreads();

    // Bin edges hoisted into registers (uniform per wave).
    const float e0 = s_edge[0], e1 = s_edge[1], e2 = s_edge[2], e3 = s_edge[3];

    const int n4 = n >> 2;
    const int gs = (int)(blockDim.x * gridDim.x);   // grid stride in float4 units
    const v4f* __restrict__ x4 = (const v4f*)x;
    v4f* __restrict__ o4       = (v4f*)out;

    int i = blockIdx.x * TPB + tid;

    // Fast path: each thread owns exactly 4 strided float4s (true for the launch
    // config when n4 % (4*gs) == 0, which holds for the reference shape).
    // Issue all 4 non-temporal 128-bit loads back-to-back for MLP.
    if (i + 3 * gs < n4) {
        v4f v0 = __builtin_nontemporal_load(&x4[i]);
        v4f v1 = __builtin_nontemporal_load(&x4[i + gs]);
        v4f v2 = __builtin_nontemporal_load(&x4[i + 2 * gs]);
        v4f v3 = __builtin_nontemporal_load(&x4[i + 3 * gs]);
        v4f r0, r1, r2, r3;
        r0.x = spline_eval5(v0.x, e0, e1, e2, e3, s_bin);
        r0.y = spline_eval5(v0.y, e0, e1, e2, e3, s_bin);
        r0.z = spline_eval5(v0.z, e0, e1, e2, e3, s_bin);
        r0.w = spline_eval5(v0.w, e0, e1, e2, e3, s_bin);
        r1.x = spline_eval5(v1.x, e0, e1, e2, e3, s_bin);
        r1.y = spline_eval5(v1.y, e0, e1, e2, e3, s_bin);
        r1.z = spline_eval5(v1.z, e0, e1, e2, e3, s_bin);
        r1.w = spline_eval5(v1.w, e0, e1, e2, e3, s_bin);
        r2.x = spline_eval5(v2.x, e0, e1, e2, e3, s_bin);
        r2.y = spline_eval5(v2.y, e0, e1, e2, e3, s_bin);
        r2.z = spline_eval5(v2.z, e0, e1, e2, e3, s_bin);
        r2.w = spline_eval5(v2.w, e0, e1, e2, e3, s_bin);
        r3.x = spline_eval5(v3.x, e0, e1, e2, e3, s_bin);
        r3.y = spline_eval5(v3.y, e0, e1, e2, e3, s_bin);
        r3.z = spline_eval5(v3.z, e0, e1, e2, e3, s_bin);
        r3.w = spline_eval5(v3.w, e0, e1, e2, e3, s_bin);
        __builtin_nontemporal_store(r0, &o4[i]);
        __builtin_nontemporal_store(r1, &o4[i + gs]);
        __builtin_nontemporal_store(r2, &o4[i + 2 * gs]);
        __builtin_nontemporal_store(r3, &o4[i + 3 * gs]);
        i += 4 * gs;
    }
    // Cleanup (generic shapes / non-exact division): grid-stride with prefetch.
    for (; i < n4; i += gs) {
        __builtin_prefetch(&x4[i + gs], 0, 0);          // global_prefetch_b8 lookahead
        v4f v = __builtin_nontemporal_load(&x4[i]);
        v4f r;
        r.x = spline_eval5(v.x, e0, e1, e2, e3, s_bin);
        r.y = spline_eval5(v.y, e0, e1, e2, e3, s_bin);
        r.z = spline_eval5(v.z, e0, e1, e2, e3, s_bin);
        r.w = spline_eval5(v.w, e0, e1, e2, e3, s_bin);
        __builtin_nontemporal_store(r, &o4[i]);
    }
    // Scalar tail (n % 4 != 0 case; never taken for the reference shape).
    for (int j = (n4 << 2) + blockIdx.x * TPB + tid; j < n; j += gs) {
        out[j] = spline_eval5(x[j], e0, e1, e2, e3, s_bin);
    }
}

// Runtime-num_bins fallback (only used if num_bins != 5).
__global__ __launch_bounds__(TPB) void tonemap_generic(
    const float* __restrict__ x, const float* __restrict__ widths,
    const float* __restrict__ heights, const float* __restrict__ slopes,
    float* __restrict__ out, int nb, int n) {
    __shared__ v4f   s_bin[2 * MAXB];
    __shared__ float s_edge[MAXB];

    const int tid = threadIdx.x;
    if (nb > MAXB) nb = MAXB;
    if (tid < nb) {
        float c = 0.0f;
        for (int k = 0; k < tid; ++k) c += widths[k];
        float w    = widths[tid];
        float xh   = c + w;
        float invw = 1.0f / ((xh - c) + 1e-8f);
        float yl   = heights[tid];
        float dy   = heights[tid + 1] - yl;
        float sl   = slopes[tid];
        float s    = sl + slopes[tid + 1];
        v4f b0 = {c, invw, yl, dy};
        v4f b1 = {sl - 2.0f, s, 2.0f * s - 4.0f, 0.0f};
        s_bin[2 * tid]     = b0;
        s_bin[2 * tid + 1] = b1;
        s_edge[tid] = xh;
    }
    __syncthreads();

    const int gs = (int)(blockDim.x * gridDim.x);
    for (int i = blockIdx.x * TPB + tid; i < n; i += gs) {
        float xf = fminf(fmaxf(x[i], 0.0f), 1.0f);
        int idx = 0;
        for (int k = 0; k < nb - 1; ++k) idx += (s_edge[k] < xf) ? 1 : 0;
        v4f b0 = s_bin[2 * idx];
        v4f b1 = s_bin[2 * idx + 1];
        float t   = (xf - b0.x) * b0.y;
        float num = t * __builtin_fmaf(b1.x, t, 2.0f);
        float den = __builtin_fmaf(t, __builtin_fmaf(b1.y, t, b1.z), 2.0f);
        float r   = __builtin_amdgcn_rcpf(den);
        r = __builtin_fmaf(__builtin_fmaf(-den, r, 1.0f), r, r);
        out[i] = __builtin_fmaf(b0.w * num, r, b0.z);
    }
}

extern "C" void kernel_launch(void* const* d_in, const int* in_sizes, int n_in,
                              void* d_out, int out_size, void* d_ws, size_t ws_size,
                              hipStream_t stream) {
    (void)n_in; (void)d_ws; (void)ws_size;
    const float* x       = (const float*)d_in[0];
    const float* widths  = (const float*)d_in[1];
    const float* heights = (const float*)d_in[2];
    const float* slopes  = (const float*)d_in[3];
    float* out = (float*)d_out;

    const int n  = out_size;          // 16*3*1024*1024 = 50,331,648
    const int nb = in_sizes[1];       // num_bins (5 for the reference)

    // 4 float4s (16 elements) per thread.
    const int n4 = n >> 2;
    long long threads_needed = ((long long)n4 + 3) / 4;
    int blocks = (int)((threads_needed + TPB - 1) / TPB);
    if (blocks < 1) blocks = 1;

    if (nb == 5) {
        tonemap5<<<blocks, TPB, 0, stream>>>(x, widths, heights, slopes, out, n);
    } else {
        tonemap_generic<<<blocks, TPB, 0, stream>>>(x, widths, heights, slopes, out, nb, n);
    }
}